// GlobalAttentionBlock_75445395521924
// MI455X (gfx1250) — compile-verified
//
#include <hip/hip_runtime.h>
#include <hip/hip_bf16.h>
#include <math.h>

#define B_   2
#define T_   2048
#define D_   1024
#define H_   16
#define HD_  64
#define STRIDE_ 8
#define MLP_ 4096
#define D3_  (3 * D_)

typedef __attribute__((ext_vector_type(2))) float v2f;
typedef __attribute__((ext_vector_type(8))) float v8f;

// D = A(16x4, f32) * B(4x16, f32) + C(16x16, f32)
__device__ __forceinline__ v8f wmma_f32(v2f a, v2f b, v8f c) {
  return __builtin_amdgcn_wmma_f32_16x16x4_f32(
      /*neg_a=*/false, a, /*neg_b=*/false, b,
      /*c_mod=*/(short)0, c, /*reuse_a=*/false, /*reuse_b=*/false);
}

// Async DMA: 16 bytes global -> LDS, tracked by ASYNCcnt (no VGPR transit).
// LDS operand is the low 32 bits of the generic pointer (flat->LDS mapping).
__device__ __forceinline__ void async_b128(const float* g, float* l) {
  unsigned lds_addr = (unsigned)(unsigned long long)l;
  asm volatile("global_load_async_to_lds_b128 %0, %1, off"
               :: "v"(lds_addr), "v"(g) : "memory");
}
__device__ __forceinline__ void wait_async_le12() {
  asm volatile("s_wait_asynccnt 0xc" ::: "memory");
}
__device__ __forceinline__ void wait_async_0() {
  asm volatile("s_wait_asynccnt 0x0" ::: "memory");
}

// ---------------------------------------------------------------------------
// Row-wise LayerNorm: one block (256 threads) per row of length D_=1024.
// ---------------------------------------------------------------------------
__global__ __launch_bounds__(256) void ln_kernel(const float* __restrict__ x,
                                                 const float* __restrict__ g,
                                                 const float* __restrict__ b,
                                                 float* __restrict__ out) {
  int row = blockIdx.x;
  int tid = threadIdx.x;
  const float* xr = x + (size_t)row * D_;
  float* orow = out + (size_t)row * D_;

  float v0 = xr[tid], v1 = xr[tid + 256], v2 = xr[tid + 512], v3 = xr[tid + 768];
  float s  = v0 + v1 + v2 + v3;
  float ss = v0 * v0 + v1 * v1 + v2 * v2 + v3 * v3;

  __shared__ float rs[256];
  __shared__ float rq[256];
  rs[tid] = s; rq[tid] = ss;
  __syncthreads();
  for (int off = 128; off > 0; off >>= 1) {
    if (tid < off) { rs[tid] += rs[tid + off]; rq[tid] += rq[tid + off]; }
    __syncthreads();
  }
  float mean = rs[0] * (1.0f / D_);
  float var  = rq[0] * (1.0f / D_) - mean * mean;
  float inv  = rsqrtf(var + 1e-5f);

  orow[tid]       = (v0 - mean) * inv * g[tid]       + b[tid];
  orow[tid + 256] = (v1 - mean) * inv * g[tid + 256] + b[tid + 256];
  orow[tid + 512] = (v2 - mean) * inv * g[tid + 512] + b[tid + 512];
  orow[tid + 768] = (v3 - mean) * inv * g[tid + 768] + b[tid + 768];
}

// ---------------------------------------------------------------------------
// GEMM: out[M,N] = act(A[M,K] * W[N,K]^T + bias[N]) + resid[M,N]
// 4 waves (2x2) per block -> 64x128 tile. K consumed in 32-deep blocks,
// double-buffered in LDS via GLOBAL_LOAD_ASYNC_TO_LDS_B128 (ASYNCcnt).
// Inner loop: 6x ds_load_b64 + 8x V_WMMA_F32_16X16X4_F32 per k-step.
// act: 0 = identity, 1 = exact GELU.
// ---------------------------------------------------------------------------
#define KB   32                 // K-block depth
#define AST  36                 // padded LDS row stride (floats): 16B-aligned
                                // rows + conflict-free b64 fragment reads

__global__ __launch_bounds__(128) void gemm_kernel(const float* __restrict__ A,
                                                   const float* __restrict__ W,
                                                   const float* __restrict__ bias,
                                                   const float* __restrict__ resid,
                                                   float* __restrict__ out,
                                                   int M, int N, int K, int act) {
  __shared__ float As[2][64 * AST];    // 2 x 9216 B
  __shared__ float Ws[2][128 * AST];   // 2 x 18432 B

  int ntn    = N >> 7;                 // tiles along N (128 wide)
  int tile   = blockIdx.x;
  int tn     = tile % ntn;
  int tm     = tile / ntn;
  int m_base = tm * 64;
  int n_base = tn * 128;

  int tid  = threadIdx.x;
  int wave = tid >> 5;
  int lane = tid & 31;
  bool lo  = lane < 16;
  int col  = lane & 15;
  int khalf = lo ? 0 : 2;
  int wm   = (wave >> 1) * 32;         // wave row offset in tile
  int wn   = (wave & 1) * 64;          // wave col offset in tile

  v8f acc[2][4] = {};

  // Cooperative async staging of one K-block (12 B128 issues per thread).
  auto stage = [&](int buf, int kb) {
#pragma unroll
    for (int i = 0; i < 4; ++i) {      // A: 64 rows x 32 k
      int seg = tid + 128 * i;
      int row = seg >> 3;
      int ks  = (seg & 7) * 4;
      async_b128(A + (size_t)(m_base + row) * K + kb + ks,
                 &As[buf][row * AST + ks]);
    }
#pragma unroll
    for (int i = 0; i < 8; ++i) {      // W: 128 rows x 32 k
      int seg = tid + 128 * i;
      int row = seg >> 3;
      int ks  = (seg & 7) * 4;
      async_b128(W + (size_t)(n_base + row) * K + kb + ks,
                 &Ws[buf][row * AST + ks]);
    }
  };

  int nblocks = K / KB;
  stage(0, 0);

  for (int blk = 0; blk < nblocks; ++blk) {
    int buf = blk & 1;
    if (blk + 1 < nblocks) {
      stage(buf ^ 1, (blk + 1) * KB);  // prefetch next block
      wait_async_le12();               // oldest 12 (this block) complete
    } else {
      wait_async_0();
    }
    __syncthreads();

    const float* as = &As[buf][0];
    const float* ws = &Ws[buf][0];
#pragma unroll
    for (int k = 0; k < KB; k += 4) {
      v2f a0 = *(const v2f*)(as + (wm + col) * AST + k + khalf);
      v2f a1 = *(const v2f*)(as + (wm + 16 + col) * AST + k + khalf);
      v2f b0 = *(const v2f*)(ws + (wn + col) * AST + k + khalf);
      v2f b1 = *(const v2f*)(ws + (wn + 16 + col) * AST + k + khalf);
      v2f b2 = *(const v2f*)(ws + (wn + 32 + col) * AST + k + khalf);
      v2f b3 = *(const v2f*)(ws + (wn + 48 + col) * AST + k + khalf);
      acc[0][0] = wmma_f32(a0, b0, acc[0][0]);
      acc[1][0] = wmma_f32(a1, b0, acc[1][0]);
      acc[0][1] = wmma_f32(a0, b1, acc[0][1]);
      acc[1][1] = wmma_f32(a1, b1, acc[1][1]);
      acc[0][2] = wmma_f32(a0, b2, acc[0][2]);
      acc[1][2] = wmma_f32(a1, b2, acc[1][2]);
      acc[0][3] = wmma_f32(a0, b3, acc[0][3]);
      acc[1][3] = wmma_f32(a1, b3, acc[1][3]);
    }
    __syncthreads();                   // protect buf before it is re-staged
  }

  int m0 = m_base + wm;
  int n0 = n_base + wn;
  int rbase = lo ? 0 : 8;
#pragma unroll
  for (int i = 0; i < 2; ++i) {
#pragma unroll
    for (int gidx = 0; gidx < 4; ++gidx) {
      int n = n0 + 16 * gidx + col;
      float bb = bias ? bias[n] : 0.0f;
#pragma unroll
      for (int r = 0; r < 8; ++r) {
        int m = m0 + 16 * i + r + rbase;
        float v = acc[i][gidx][r] + bb;
        if (act == 1) v = 0.5f * v * (1.0f + erff(v * 0.70710678118654752f));
        if (resid) v += resid[(size_t)m * N + n];
        out[(size_t)m * N + n] = v;
      }
    }
  }
}

// ---------------------------------------------------------------------------
// Non-stride rows: attention collapses to self -> o = v.
// ---------------------------------------------------------------------------
__global__ __launch_bounds__(256) void vcopy_kernel(const float* __restrict__ qkv,
                                                    float* __restrict__ o) {
  size_t idx = (size_t)blockIdx.x * 256 + threadIdx.x;   // over B*T*D
  size_t row = idx / D_;
  int colD   = (int)(idx % D_);
  int t      = (int)(row % T_);
  if ((t & (STRIDE_ - 1)) != 0) {
    o[idx] = qkv[row * D3_ + 2 * D_ + colD];
  }
}

// ---------------------------------------------------------------------------
// Full attention for the 256 stride rows per (b,h).
// One wave per (b, h, 16 stride queries). Scores staged in 131KB dynamic LDS.
//   S^T = K * Q^T  (WMMA, key-major so softmax is per-lane)
//   softmax rows in LDS
//   O   = P * V    (WMMA, A-fragments read straight from LDS)
// ---------------------------------------------------------------------------
#define SROW 2052   // 2048 keys + pad to kill bank conflicts

__global__ __launch_bounds__(32) void attn_kernel(const float* __restrict__ qkv,
                                                  float* __restrict__ o) {
  extern __shared__ float sS[];     // [16 queries][SROW keys]

  int wg  = blockIdx.x;             // B*H*16
  int qt  = wg & 15;
  int h   = (wg >> 4) & 15;
  int b   = wg >> 8;
  int lane = threadIdx.x;
  bool lo  = lane < 16;
  int col  = lane & 15;
  int khalf = lo ? 0 : 2;
  const float* base = qkv + (size_t)b * T_ * D3_;

  // Preload Q fragments as the B operand of S^T = K * Q^T, pre-scaled by 1/8.
  v2f qf[16];
  {
    int tq = qt * 128 + 8 * col;                       // query row for lane's column
    const float* qp = base + (size_t)tq * D3_ + h * HD_ + khalf;
#pragma unroll
    for (int kk = 0; kk < 16; ++kk) {
      v2f t = *(const v2f*)(qp + 4 * kk);
      qf[kk].x = t.x * 0.125f;
      qf[kk].y = t.y * 0.125f;
    }
  }

  // Phase 1: scores S^T (keys x queries), 16 keys per chunk.
  for (int c = 0; c < 128; ++c) {
    v8f s = {};
    int tk = c * 16 + col;                             // key row for A fragment
    const float* kp = base + (size_t)tk * D3_ + D_ + h * HD_ + khalf;
#pragma unroll
    for (int kk = 0; kk < 16; ++kk) {
      v2f af = *(const v2f*)(kp + 4 * kk);
      s = wmma_f32(af, qf[kk], s);
    }
#pragma unroll
    for (int r = 0; r < 8; ++r) {
      int key = c * 16 + r + (lo ? 0 : 8);
      sS[col * SROW + key] = s[r];                     // store as [query][key]
    }
  }
  __syncthreads();

  // Phase 2: softmax along keys. Lane handles query `col`, half the keys.
  float inv;
  {
    int koff = lo ? 0 : 1024;
    float* p = sS + col * SROW + koff;
    float mx = -3.4e38f;
    for (int k2 = 0; k2 < 1024; ++k2) mx = fmaxf(mx, p[k2]);
    mx = fmaxf(mx, __shfl_xor(mx, 16, 32));
    float sum = 0.0f;
    for (int k2 = 0; k2 < 1024; ++k2) {
      float e = __expf(p[k2] - mx);
      p[k2] = e;
      sum += e;
    }
    sum += __shfl_xor(sum, 16, 32);
    inv = 1.0f / sum;                                  // per query = per lane col
  }
  __syncthreads();

  // Phase 3: O = P * V, accumulating 16x64 across 128 key chunks.
  v8f oa[4] = {};
  for (int c = 0; c < 128; ++c) {
#pragma unroll
    for (int kk2 = 0; kk2 < 4; ++kk2) {
      int key0 = c * 16 + 4 * kk2 + khalf;
      v2f af = *(const v2f*)(sS + col * SROW + key0);  // P in A layout, free
      const float* vp = base + (size_t)key0 * D3_ + 2 * D_ + h * HD_ + col;
#pragma unroll
      for (int g = 0; g < 4; ++g) {
        v2f bf;
        bf.x = vp[16 * g];                             // V[key0  ][d]
        bf.y = vp[(size_t)D3_ + 16 * g];               // V[key0+1][d]
        oa[g] = wmma_f32(af, bf, oa[g]);
      }
    }
  }

  // Epilogue: scale rows by 1/sum and scatter to o.
  float* ob = o + (size_t)b * T_ * D_;
#pragma unroll
  for (int r = 0; r < 8; ++r) {
    int q = r + (lo ? 0 : 8);
    float invq = __shfl(inv, q, 32);
    int tq2 = qt * 128 + 8 * q;
#pragma unroll
    for (int g = 0; g < 4; ++g) {
      ob[(size_t)tq2 * D_ + h * HD_ + 16 * g + col] = oa[g][r] * invq;
    }
  }
}

// ---------------------------------------------------------------------------
extern "C" void kernel_launch(void* const* d_in, const int* in_sizes, int n_in,
                              void* d_out, int out_size, void* d_ws, size_t ws_size,
                              hipStream_t stream) {
  (void)in_sizes; (void)n_in; (void)out_size; (void)ws_size;

  const float* x     = (const float*)d_in[0];
  const float* w_in  = (const float*)d_in[1];
  const float* b_in  = (const float*)d_in[2];
  const float* w_out = (const float*)d_in[3];
  const float* b_out = (const float*)d_in[4];
  const float* w1    = (const float*)d_in[5];
  const float* b1    = (const float*)d_in[6];
  const float* w2    = (const float*)d_in[7];
  const float* b2    = (const float*)d_in[8];
  const float* ln1g  = (const float*)d_in[9];
  const float* ln1b  = (const float*)d_in[10];
  const float* ln2g  = (const float*)d_in[11];
  const float* ln2b  = (const float*)d_in[12];

  char* ws = (char*)d_ws;
  float* xn   = (float*)(ws);                               // 16 MB
  float* qkv  = (float*)(ws + (size_t)16  * 1024 * 1024);   // 48 MB
  float* ob   = (float*)(ws + (size_t)64  * 1024 * 1024);   // 16 MB
  float* x1   = (float*)(ws + (size_t)80  * 1024 * 1024);   // 16 MB
  float* hbuf = (float*)(ws + (size_t)96  * 1024 * 1024);   // 64 MB
  float* outp = (float*)d_out;

  const int M = B_ * T_;                                    // 4096 rows

  // xn = LN1(x)
  ln_kernel<<<M, 256, 0, stream>>>(x, ln1g, ln1b, xn);
  // qkv = xn @ w_in^T + b_in
  gemm_kernel<<<(M / 64) * (D3_ / 128), 128, 0, stream>>>(xn, w_in, b_in, nullptr,
                                                          qkv, M, D3_, D_, 0);
  // non-stride rows: o = v
  vcopy_kernel<<<(M * D_) / 256, 256, 0, stream>>>(qkv, ob);
  // stride rows: full masked attention
  attn_kernel<<<B_ * H_ * 16, 32, 16 * SROW * sizeof(float), stream>>>(qkv, ob);
  // x1 = x + o @ w_out^T + b_out
  gemm_kernel<<<(M / 64) * (D_ / 128), 128, 0, stream>>>(ob, w_out, b_out, x,
                                                         x1, M, D_, D_, 0);
  // xn = LN2(x1)
  ln_kernel<<<M, 256, 0, stream>>>(x1, ln2g, ln2b, xn);
  // h = gelu(xn @ w1^T + b1)
  gemm_kernel<<<(M / 64) * (MLP_ / 128), 128, 0, stream>>>(xn, w1, b1, nullptr,
                                                           hbuf, M, MLP_, D_, 1);
  // out = x1 + h @ w2^T + b2
  gemm_kernel<<<(M / 64) * (D_ / 128), 128, 0, stream>>>(hbuf, w2, b2, x1,
                                                         outp, M, D_, MLP_, 0);
}